// PSRoIPoolAfterPointwiseConv_16819091931426
// MI455X (gfx1250) — compile-verified
//
#include <hip/hip_runtime.h>
#include <hip/hip_bf16.h>

// Problem constants (from reference setup_inputs)
#define BB   4      // batch
#define CIN  1024   // input channels
#define HH   64
#define WW   64
#define COUT 1029
#define DD   21     // COUT / 49
#define GG   7
#define NROI 512
#define NBIN 49     // G*G
#define KC   128    // K-chunk for the fused pool+GEMM kernel

typedef __attribute__((ext_vector_type(2))) float v2f;
typedef __attribute__((ext_vector_type(4))) float v4f;
typedef __attribute__((ext_vector_type(8))) float v8f;

// ---------------------------------------------------------------------------
// Stage 1: row-wise inclusive cumsum over W, transposed to channel-last
// R[b][h][w][c] = sum_{w'<=w} feat[b][c][h][w']
// Block: 256 threads; one (b, h, 32-channel tile) per block; LDS transpose.
// ---------------------------------------------------------------------------
__global__ void k_rowscan_transpose(const float* __restrict__ feat,
                                    float* __restrict__ R) {
    __shared__ float tile[32][WW + 1];   // +1 pad -> conflict-free scans
    const int bid   = blockIdx.x;
    const int ctile = bid & 31;          // 32 tiles of 32 channels
    const int h     = (bid >> 5) & 63;
    const int b     = bid >> 11;
    const int tid   = threadIdx.x;

    const float* src = feat + (((size_t)b * CIN + (size_t)ctile * 32) * HH + h) * WW;
#pragma unroll
    for (int i = 0; i < 8; ++i) {        // coalesced: 64-float rows per channel
        int idx = tid + i * 256;
        int cl  = idx >> 6;
        int w   = idx & 63;
        tile[cl][w] = src[(size_t)cl * (HH * WW) + w];
    }
    __syncthreads();

    if (tid < 32) {                       // serial scan per channel, reg accumulator
        float s = 0.0f;
#pragma unroll
        for (int w = 0; w < WW; ++w) { s += tile[tid][w]; tile[tid][w] = s; }
    }
    __syncthreads();

    float* dst = R + ((size_t)b * HH + h) * WW * CIN + (size_t)ctile * 32;
#pragma unroll
    for (int i = 0; i < 8; ++i) {        // coalesced 128B segments per w
        int idx = tid + i * 256;
        int w   = idx >> 5;
        int cl  = idx & 31;
        dst[(size_t)w * CIN + cl] = tile[cl][w];
    }
}

// ---------------------------------------------------------------------------
// Stage 2: column cumsum over H, in place on R. One thread per (b, w, c4)
// handles 4 consecutive channels with b128 loads/stores (coalesced).
// ---------------------------------------------------------------------------
__global__ void k_colscan(float* __restrict__ R) {
    const int g  = blockIdx.x * blockDim.x + threadIdx.x;   // B*W*(C/4) threads
    const int c4 = g & 255;                                 // 256 float4 per row
    const int w  = (g >> 8) & 63;
    const int b  = g >> 14;
    const size_t hstride = (size_t)WW * CIN;
    const size_t base = ((size_t)b * HH * WW + w) * CIN + (size_t)c4 * 4;  // h=0
    v4f s = *(const v4f*)(R + base);
    for (int h = 1; h < HH; ++h) {
        v4f* p = (v4f*)(R + base + (size_t)h * hstride);
        s += *p;
        *p = s;
    }
}

// ---------------------------------------------------------------------------
// Fused Stage 3+4: PSRoI pooling (4 integral-image corners) directly into an
// LDS A-tile, consumed by V_WMMA_F32_16X16X4_F32, with fused BN + empty-bin
// mask epilogue. Block = 256 threads = 8 waves; grid = 49 bins x 8 chunks of
// 64 ROIs. K is processed in chunks of KC=128:
//   producer: all threads gather corners (coalesced float4 per corner, since
//             corner offsets depend only on the ROI row) -> T[64][KC]
//   consumer: each wave runs 32 WMMAs on its 16(roi) x 16(D-chan) tile.
// LDS row stride = KC+4 = 132 words => banks 4*row mod 64: the 16 rows of a
// WMMA A-fragment + the two K-halves tile all 64 banks exactly once.
// ---------------------------------------------------------------------------
__global__ void k_pool_gemm(const float* __restrict__ R,
                            const float* __restrict__ rois,
                            const float* __restrict__ cw,
                            const float* __restrict__ gamma,
                            const float* __restrict__ beta,
                            const float* __restrict__ mean,
                            const float* __restrict__ var,
                            float* __restrict__ out) {
    __shared__ float T[64][KC + 4];
    __shared__ int   offs[64][4];
    __shared__ float wgts[64][4];
    __shared__ float msk[64];

    const int bin    = blockIdx.x % NBIN;
    const int nchunk = blockIdx.x / NBIN;
    const int nbase0 = nchunk * 64;
    const int tid    = threadIdx.x;

    // --- per-ROI geometry: corner offsets + signed weights (+-1/area) ---
    if (tid < 64) {
        const int n = nbase0 + tid;
        const int gh = bin / GG, gw = bin - gh * GG;
        const float* r = rois + (size_t)n * 5;
        const int   b  = (int)r[0];
        const float x1 = nearbyintf(r[1]) * 0.0625f;
        const float y1 = nearbyintf(r[2]) * 0.0625f;
        const float x2 = (nearbyintf(r[3]) + 1.0f) * 0.0625f;
        const float y2 = (nearbyintf(r[4]) + 1.0f) * 0.0625f;
        const float bin_w = fmaxf(x2 - x1, 0.1f) * (1.0f / (float)GG);
        const float bin_h = fmaxf(y2 - y1, 0.1f) * (1.0f / (float)GG);
        int hs = (int)floorf((float)gh * bin_h + y1);
        int he = (int)ceilf(((float)gh + 1.0f) * bin_h + y1);
        int ws = (int)floorf((float)gw * bin_w + x1);
        int we = (int)ceilf(((float)gw + 1.0f) * bin_w + x1);
        hs = min(max(hs, 0), HH); he = min(max(he, 0), HH);
        ws = min(max(ws, 0), WW); we = min(max(we, 0), WW);
        const int   area = (he - hs) * (we - ws);
        const float inv  = (area > 0) ? (1.0f / (float)area) : 0.0f;
        msk[tid] = (area > 0) ? 1.0f : 0.0f;
        const int   hc[4] = { he,  hs,   he,  hs };
        const int   wc[4] = { we,  we,   ws,  ws };
        const float sg[4] = { inv, -inv, -inv, inv };
#pragma unroll
        for (int j = 0; j < 4; ++j) {
            const bool valid = (hc[j] > 0) && (wc[j] > 0);
            offs[tid][j] = valid ? ((b * HH * WW) + (hc[j] - 1) * WW + (wc[j] - 1)) * CIN : 0;
            wgts[tid][j] = valid ? sg[j] : 0.0f;
        }
    }
    __syncthreads();

    // --- wave decomposition for the WMMA consumer ---
    const int wid   = tid >> 5;          // 8 waves (wave32)
    const int lane  = tid & 31;
    const int ntile = wid >> 1;          // 4 roi tiles of 16
    const int ctile = wid & 1;           // 2 column tiles (D=21 -> 16 + 5)
    const int lo    = lane & 15;
    const int hi    = lane >> 4;
    const int col   = ctile * 16 + lo;          // D-channel (valid < 21)
    const int o     = col * NBIN + bin;         // conv output channel
    const int oc    = min(o, COUT - 1);         // keep padded lanes in-bounds
    const float* Bptr = cw + (size_t)oc * CIN + 2 * hi;
    const float* Arow = &T[ntile * 16 + lo][2 * hi];

    v8f acc = {};
    for (int kc = 0; kc < CIN; kc += KC) {
        // producer: 64 rows x 32 float4 = 2048 elems, 8 per thread.
        // A wave covers one row's 32 float4s -> 512B coalesced per corner.
#pragma unroll
        for (int i = 0; i < 8; ++i) {
            const int idx = tid + i * 256;
            const int row = idx >> 5;
            const int c4  = idx & 31;
            const int cb  = kc + c4 * 4;
            v4f v = wgts[row][0] * (*(const v4f*)(R + offs[row][0] + cb))
                  + wgts[row][1] * (*(const v4f*)(R + offs[row][1] + cb))
                  + wgts[row][2] * (*(const v4f*)(R + offs[row][2] + cb))
                  + wgts[row][3] * (*(const v4f*)(R + offs[row][3] + cb));
            *(v4f*)&T[row][c4 * 4] = v;
        }
        __syncthreads();

        // consumer: 32 WMMAs on this K-chunk (A from LDS, B from global/L2)
#pragma unroll 8
        for (int kk = 0; kk < KC; kk += 4) {
            v2f a  = *(const v2f*)(Arow + kk);
            v2f bv = *(const v2f*)(Bptr + kc + kk);
            acc = __builtin_amdgcn_wmma_f32_16x16x4_f32(
                /*neg_a=*/false, a, /*neg_b=*/false, bv,
                /*c_mod=*/(short)0, acc, /*reuse_a=*/false, /*reuse_b=*/false);
        }
        __syncthreads();
    }

    // --- epilogue: BN affine + empty-bin mask ---
    if (col < DD) {
        const float sc = gamma[o] * rsqrtf(var[o] + 1e-5f);
        const float sh = beta[o] - mean[o] * sc;
#pragma unroll
        for (int rr = 0; rr < 8; ++rr) {
            const int row = ntile * 16 + rr + 8 * hi;    // C/D layout: VGPR rr -> M=rr(+8)
            const int n   = nbase0 + row;
            const float v = (msk[row] != 0.0f) ? (acc[rr] * sc + sh) : 0.0f;
            out[((size_t)n * DD + col) * NBIN + bin] = v;
        }
    }
}

// ---------------------------------------------------------------------------
extern "C" void kernel_launch(void* const* d_in, const int* in_sizes, int n_in,
                              void* d_out, int out_size, void* d_ws, size_t ws_size,
                              hipStream_t stream) {
    (void)in_sizes; (void)n_in; (void)out_size; (void)ws_size;

    const float* feat  = (const float*)d_in[0];   // [4,1024,64,64]
    const float* rois  = (const float*)d_in[1];   // [512,5]
    const float* cw    = (const float*)d_in[2];   // [1029,1024]
    const float* gamma = (const float*)d_in[3];
    const float* beta  = (const float*)d_in[4];
    const float* mean  = (const float*)d_in[5];
    const float* var   = (const float*)d_in[6];
    float* out = (float*)d_out;                   // [512,21,7,7]

    float* R = (float*)d_ws;                      // 67 MB integral image (only ws use)

    k_rowscan_transpose<<<BB * HH * 32, 256, 0, stream>>>(feat, R);
    k_colscan<<<(BB * WW * (CIN / 4)) / 256, 256, 0, stream>>>(R);
    k_pool_gemm<<<NBIN * (NROI / 64), 256, 0, stream>>>(R, rois, cw, gamma, beta,
                                                        mean, var, out);
}